// CASSI_41721312314035
// MI455X (gfx1250) — compile-verified
//
#include <hip/hip_runtime.h>
#include <stdint.h>

// Problem constants (B=1): x:(1,31,1024,1024) f32, ca:(1,1,1024,1024) f32
// out:(1,1,1024,1054) f32 ; out[m][j] = sum_l x[l][m][j-l]*ca[m][j-l]
#define L_BANDS 31
#define MROWS   1024
#define NCOLS   1024
#define NOUT    (NCOLS + L_BANDS - 1)   // 1054
#define NOUT_PAD 1280                   // 5*256: uniform zeroing, full EXEC
#define TPB     256                     // 8 wave32 per block
#define NBUF    4                       // ring of row buffers, prefetch dist 2

// CDNA5 async DMA: copy 16 B per lane from global to LDS, tracked by ASYNCcnt.
// vdst = LDS byte offset (VGPR), vaddr = 64-bit global address (VGPR pair).
__device__ __forceinline__ void async_b128_to_lds(uint32_t lds_off_bytes,
                                                  const float* gsrc) {
    uint64_t gaddr = (uint64_t)(uintptr_t)gsrc;
    asm volatile("global_load_async_to_lds_b128 %0, %1, off"
                 :: "v"(lds_off_bytes), "v"(gaddr)
                 : "memory");
}

template <int N>
__device__ __forceinline__ void wait_async() {
#if defined(__has_builtin) && __has_builtin(__builtin_amdgcn_s_wait_asynccnt)
    __builtin_amdgcn_s_wait_asynccnt(N);
#else
    if constexpr (N == 0) asm volatile("s_wait_asynccnt 0x0" ::: "memory");
    else                  asm volatile("s_wait_asynccnt 0x1" ::: "memory");
#endif
}

__global__ __launch_bounds__(TPB)
void cassi_fwd_kernel(const float* __restrict__ x,
                      const float* __restrict__ ca,
                      float* __restrict__ out) {
    // 16B alignment required: b128 async writes land on 16B LDS chunks.
    __shared__ __align__(16) float s_ca[NCOLS];          // 4 KB
    __shared__ __align__(16) float s_x[NBUF][NCOLS];     // 16 KB ring buffer
    __shared__ __align__(16) float s_out[NOUT_PAD];      // 5 KB accumulator

    const int m   = blockIdx.x;
    const int tid = threadIdx.x;

    const float* ca_row = ca + (size_t)m * NCOLS;
    const float* x_row0 = x  + (size_t)m * NCOLS;        // band 0, row m
    const size_t l_step = (size_t)MROWS * NCOLS;         // stride between bands

    // LDS byte offsets of this lane's 16 B slice (256 lanes * 16 B = one row).
    const uint32_t lane16  = (uint32_t)tid * 16u;
    const uint32_t lds_ca  = (uint32_t)(uintptr_t)(&s_ca[0])   + lane16;
    const uint32_t lds_xb  = (uint32_t)(uintptr_t)(&s_x[0][0]) + lane16;
    // buffer b lives at lds_xb + b*4096

    // Prologue: stage ca row, then bands 0 and 1 (in-order ASYNCcnt retire).
    async_b128_to_lds(lds_ca,                 ca_row + tid * 4);
    async_b128_to_lds(lds_xb,                 x_row0 + tid * 4);
    async_b128_to_lds(lds_xb + 4096u,         x_row0 + l_step + tid * 4);

    // Zero the accumulator row (uniform, full EXEC) while the DMA flies.
    #pragma unroll
    for (int k = 0; k < NOUT_PAD / TPB; ++k) s_out[tid + k * TPB] = 0.0f;

    wait_async<1>();        // ca + band0 retired; band1 may still be in flight
    __syncthreads();        // all waves' slices visible; s_out zeroed

    // ca values this thread uses are loop-invariant: cache in registers.
    float rca[NCOLS / TPB];
    #pragma unroll
    for (int k = 0; k < NCOLS / TPB; ++k) rca[k] = s_ca[tid + k * TPB];

    #pragma unroll 1
    for (int l = 0; l < L_BANDS; ++l) {
        // Prefetch band l+2 into the ring (its previous tenant, band l-2,
        // was last read before the barrier that ended iteration l-2).
        if (l + 2 < L_BANDS) {
            async_b128_to_lds(lds_xb + (uint32_t)((l + 2) & (NBUF - 1)) * 4096u,
                              x_row0 + (size_t)(l + 2) * l_step + tid * 4);
        }

        // Accumulate band l: element c+l is owned by exactly one thread this
        // iteration, so plain LDS RMW is race-free; barriers order iterations.
        const float* sx = s_x[l & (NBUF - 1)];
        #pragma unroll
        for (int k = 0; k < NCOLS / TPB; ++k) {
            const int c = tid + k * TPB;
            s_out[c + l] += sx[c] * rca[k];
        }

        // Retire band l+1 (leave l+2 in flight when it exists), then make it
        // visible to the whole workgroup.
        if (l + 2 < L_BANDS) wait_async<1>();
        else                 wait_async<0>();
        __syncthreads();
    }

    // Coalesced write-out of the 1054-column measurement row.
    float* out_row = out + (size_t)m * NOUT;
    for (int j = tid; j < NOUT; j += TPB) out_row[j] = s_out[j];
}

extern "C" void kernel_launch(void* const* d_in, const int* in_sizes, int n_in,
                              void* d_out, int out_size, void* d_ws, size_t ws_size,
                              hipStream_t stream) {
    (void)in_sizes; (void)n_in; (void)out_size; (void)d_ws; (void)ws_size;
    const float* x  = (const float*)d_in[0];   // (1,31,1024,1024)
    const float* ca = (const float*)d_in[1];   // (1,1,1024,1024)
    float* out      = (float*)d_out;           // (1,1,1024,1054)
    cassi_fwd_kernel<<<dim3(MROWS), dim3(TPB), 0, stream>>>(x, ca, out);
}